// SparseAttention_6700148982484
// MI455X (gfx1250) — compile-verified
//
#include <hip/hip_runtime.h>

typedef float v2f __attribute__((ext_vector_type(2)));
typedef float v4f __attribute__((ext_vector_type(4)));
typedef float v8f __attribute__((ext_vector_type(8)));

constexpr int kS  = 2048;   // sequence length (queries == keys)
constexpr int kD  = 128;    // head dim
constexpr int kTQ = 16;     // queries per workgroup
constexpr int kWaves = 8;   // 256 threads, wave32
constexpr int kKeysPerWave = kS / kWaves;  // 256
constexpr int kNIter = 100;

// LDS: scores/weights [16][2048] f32 (128KB) + V stage [32][128] f32 (16KB)
constexpr int kVStageOffBytes = kTQ * kS * 4;      // 131072
constexpr int kLdsFloats = kTQ * kS + 32 * kD;

__global__ __launch_bounds__(256)
void entmax15_attn_f32(const float* __restrict__ Qg,
                       const float* __restrict__ Kg,
                       const float* __restrict__ Vg,
                       float* __restrict__ Og) {
  extern __shared__ float lds[];
  float* sc     = lds;              // [kTQ][kS] scores -> entmax weights
  float* vstage = lds + kTQ * kS;   // [32][kD]

  const int b    = blockIdx.x / (kS / kTQ);
  const int qb   = (blockIdx.x % (kS / kTQ)) * kTQ;
  const int tid  = threadIdx.x;
  const int wave = tid >> 5;
  const int lane = tid & 31;
  const int hf   = lane >> 4;   // which half of the wave (K-offset selector)
  const int l16  = lane & 15;   // row (A) / col (B,C,D) index

  const float* Qb = Qg + ((size_t)b * kS + qb) * kD;
  const float* Kb = Kg + (size_t)b * kS * kD;
  const float* Vb = Vg + (size_t)b * kS * kD;

  // ---------------- Phase 1: scores = Q K^T  (V_WMMA_F32_16X16X4_F32) -------
  // f32 16x4 A layout: lane m(<16) holds A[m, k0..k0+1]; lane m+16 holds A[m, k0+2..k0+3].
  // f32 4x16 B layout is lane-symmetric with columns, so K loads identically.
  v2f qreg[kD / 4];
  {
    const float* qp = Qb + l16 * kD + hf * 2;
    #pragma unroll
    for (int j = 0; j < kD / 4; ++j)
      qreg[j] = *(const v2f*)(qp + 4 * j);
  }

  #pragma unroll 1
  for (int t = 0; t < kKeysPerWave / 16; ++t) {
    const int key0 = wave * kKeysPerWave + t * 16;
    const float* kp = Kb + (size_t)(key0 + l16) * kD + hf * 2;
    // two independent accumulator chains to halve XDL RAW-dependency depth
    v8f acc0 = {0.f, 0.f, 0.f, 0.f, 0.f, 0.f, 0.f, 0.f};
    v8f acc1 = {0.f, 0.f, 0.f, 0.f, 0.f, 0.f, 0.f, 0.f};
    #pragma unroll
    for (int j = 0; j < kD / 4; j += 2) {
      v2f bf0 = *(const v2f*)(kp + 4 * j);
      v2f bf1 = *(const v2f*)(kp + 4 * (j + 1));
      acc0 = __builtin_amdgcn_wmma_f32_16x16x4_f32(false, qreg[j], false, bf0,
                                                   (short)0, acc0, false, false);
      acc1 = __builtin_amdgcn_wmma_f32_16x16x4_f32(false, qreg[j + 1], false, bf1,
                                                   (short)0, acc1, false, false);
    }
    // C layout: VGPR r -> M = r + 8*hf, N = l16
    #pragma unroll
    for (int r = 0; r < 8; ++r)
      sc[(r + 8 * hf) * kS + key0 + l16] = acc0[r] + acc1[r];
  }
  __syncthreads();

  // ---------------- Phase 2: entmax (alpha=1.5) bisection, rows in registers -
  // wave w owns rows 2w, 2w+1; lane holds float4 chunks k = 4*lane + 128*j
  v4f zreg[2][16];
  float tauv[2], Zv[2];
  {
    float tmin[2], tmax[2];
    #pragma unroll
    for (int r = 0; r < 2; ++r) {
      const int row = 2 * wave + r;
      float m = -3.402823466e38f;
      #pragma unroll
      for (int j = 0; j < 16; ++j) {
        v4f tz = *(const v4f*)&sc[row * kS + 4 * lane + 128 * j];
        tz = tz * 0.5f;                      // z = (alpha-1) * scores
        zreg[r][j] = tz;
        m = fmaxf(m, fmaxf(fmaxf(tz[0], tz[1]), fmaxf(tz[2], tz[3])));
      }
      #pragma unroll
      for (int off = 16; off > 0; off >>= 1)
        m = fmaxf(m, __shfl_xor(m, off, 32));
      tmin[r] = m - 1.0f;
      tmax[r] = m - 0.022097087f;            // zmax - S^(1-alpha) = zmax - 2048^-0.5
    }

    #pragma unroll 1
    for (int it = 0; it < kNIter; ++it) {
      #pragma unroll
      for (int r = 0; r < 2; ++r) {
        const float ta = 0.5f * (tmin[r] + tmax[r]);
        float s = 0.0f;
        #pragma unroll
        for (int j = 0; j < 16; ++j) {
          v4f d = zreg[r][j];
          float a0 = fmaxf(d[0] - ta, 0.0f);
          float a1 = fmaxf(d[1] - ta, 0.0f);
          float a2 = fmaxf(d[2] - ta, 0.0f);
          float a3 = fmaxf(d[3] - ta, 0.0f);
          s = fmaf(a0, a0, s); s = fmaf(a1, a1, s);
          s = fmaf(a2, a2, s); s = fmaf(a3, a3, s);
        }
        #pragma unroll
        for (int off = 16; off > 0; off >>= 1)
          s += __shfl_xor(s, off, 32);
        // uniform select: no EXEC divergence
        tmin[r] = (s >= 1.0f) ? ta : tmin[r];
        tmax[r] = (s <  1.0f) ? ta : tmax[r];
        tauv[r] = ta;    // reference uses LAST iteration's tau
        Zv[r]   = s;     // and its Z
      }
    }

    // write normalized weights back over scores in LDS
    #pragma unroll
    for (int r = 0; r < 2; ++r) {
      const int row = 2 * wave + r;
      const float ta  = tauv[r];
      const float inv = 1.0f / Zv[r];
      #pragma unroll
      for (int j = 0; j < 16; ++j) {
        v4f d = zreg[r][j];
        v4f w;
        float a0 = fmaxf(d[0] - ta, 0.0f);
        float a1 = fmaxf(d[1] - ta, 0.0f);
        float a2 = fmaxf(d[2] - ta, 0.0f);
        float a3 = fmaxf(d[3] - ta, 0.0f);
        w[0] = a0 * a0 * inv; w[1] = a1 * a1 * inv;
        w[2] = a2 * a2 * inv; w[3] = a3 * a3 * inv;
        *(v4f*)&sc[row * kS + 4 * lane + 128 * j] = w;
      }
    }
  }
  __syncthreads();

  // ---------------- Phase 3: out = P V, V staged via async global->LDS -------
  const int dt = wave * 16;   // each wave owns a 16-wide d-tile
  v8f oacc0 = {0.f, 0.f, 0.f, 0.f, 0.f, 0.f, 0.f, 0.f};
  v8f oacc1 = {0.f, 0.f, 0.f, 0.f, 0.f, 0.f, 0.f, 0.f};
  #pragma unroll 1
  for (int kb = 0; kb < kS / 32; ++kb) {
    __syncthreads();  // previous block's readers done before restaging
    {
      // GLOBAL_LOAD_ASYNC_TO_LDS_B128: per-lane 16B global -> LDS, ASYNCcnt.
      // vdst = LDS byte offset (relative to wave LDS base), vaddr = 64-bit addr.
      const char* src = (const char*)(Vb + (size_t)kb * 32 * kD);
      #pragma unroll
      for (int u = 0; u < 4; ++u) {
        const int idx = tid + 256 * u;                 // float4 units, 1024 total
        unsigned int ldsOff = kVStageOffBytes + 16u * (unsigned)idx;
        unsigned long long gaddr = (unsigned long long)(src + 16u * (unsigned)idx);
        asm volatile("global_load_async_to_lds_b128 %0, %1, off"
                     :: "v"(ldsOff), "v"(gaddr)
                     : "memory");
      }
      asm volatile("s_wait_asynccnt 0" ::: "memory");
    }
    __syncthreads();
    #pragma unroll
    for (int kk = 0; kk < 8; kk += 2) {
      const int k0 = kk * 4 + hf * 2;
      const int k1 = (kk + 1) * 4 + hf * 2;
      v2f af0 = *(const v2f*)&sc[l16 * kS + kb * 32 + k0];   // P[l16][k0..k0+1]
      v2f af1 = *(const v2f*)&sc[l16 * kS + kb * 32 + k1];
      v2f bf0, bf1;
      bf0[0] = vstage[k0 * kD + dt + l16];                   // V[k0][d]
      bf0[1] = vstage[(k0 + 1) * kD + dt + l16];             // V[k0+1][d]
      bf1[0] = vstage[k1 * kD + dt + l16];
      bf1[1] = vstage[(k1 + 1) * kD + dt + l16];
      oacc0 = __builtin_amdgcn_wmma_f32_16x16x4_f32(false, af0, false, bf0,
                                                    (short)0, oacc0, false, false);
      oacc1 = __builtin_amdgcn_wmma_f32_16x16x4_f32(false, af1, false, bf1,
                                                    (short)0, oacc1, false, false);
    }
  }

  float* Ob = Og + ((size_t)b * kS + qb) * kD + dt;
  #pragma unroll
  for (int r = 0; r < 8; ++r)
    Ob[(size_t)(r + 8 * hf) * kD + l16] = oacc0[r] + oacc1[r];
}

extern "C" void kernel_launch(void* const* d_in, const int* in_sizes, int n_in,
                              void* d_out, int out_size, void* d_ws, size_t ws_size,
                              hipStream_t stream) {
  const float* Q = (const float*)d_in[0];
  const float* K = (const float*)d_in[1];
  const float* V = (const float*)d_in[2];
  float* O = (float*)d_out;

  const int Btot = in_sizes[0] / (kS * kD);   // = 8
  const size_t shmem = (size_t)kLdsFloats * sizeof(float);  // 147456 B

  hipFuncSetAttribute(reinterpret_cast<const void*>(entmax15_attn_f32),
                      hipFuncAttributeMaxDynamicSharedMemorySize, (int)shmem);

  dim3 grid(Btot * (kS / kTQ));   // 1024 workgroups
  entmax15_attn_f32<<<grid, dim3(256), shmem, stream>>>(Q, K, V, O);
}